// HdcLevelEncoder_89773406421003
// MI455X (gfx1250) — compile-verified
//
#include <hip/hip_runtime.h>
#include <math.h>

#define LEVELS   100
#define TSTAMPS  2048
#define DIM      10000
#define NSAMP    2048
#define NSLICE   16
#define SPS      (NSAMP / NSLICE)   // 128 samples per slice

// workspace layout (bytes)
#define SIDX_OFF     0
#define SIDX_BYTES   (NSAMP * 9 * 4)          // 73728
#define EIDX_OFF     SIDX_BYTES
#define PARTIAL_OFF  (SIDX_BYTES + 512)        // 74240, 16B aligned
// partial: NSLICE * DIM floats = 640000 bytes

typedef __attribute__((ext_vector_type(2))) float v2f;
typedef __attribute__((ext_vector_type(8))) float v8f;

__device__ __forceinline__ int lvl_idx(float x, float lo, float hi, int n) {
    float f = (x - lo) / (hi - lo) * (float)(n - 1);
    f = rintf(f);                         // v_rndne: round-half-to-even == jnp.round
    f = fminf(fmaxf(f, 0.0f), (float)(n - 1));
    return (int)f;
}

// ---------------- Kernel 1: index + energy precompute ----------------
__global__ void hdc_prep(const float* __restrict__ inp,
                         int* __restrict__ sidx, int* __restrict__ eidx) {
    __shared__ float red[3][256];
    const int tid = threadIdx.x;
    const float t0 = inp[0];
    float ex = 0.f, ey = 0.f, ez = 0.f;

    for (int n = tid; n < NSAMP; n += 256) {
        float tn = inp[n * 4 + 0] - t0;
        float x = inp[n * 4 + 1];
        float y = inp[n * 4 + 2];
        float z = inp[n * 4 + 3];
        ex += x * x; ey += y * y; ez += z * z;

        int ix = lvl_idx(x, -5.f, 5.f, LEVELS);
        int iy = lvl_idx(y, -5.f, 5.f, LEVELS);
        int iz = lvl_idx(z, -5.f, 5.f, LEVELS);
        float mag = sqrtf(x * x + y * y + z * z);
        int im = lvl_idx(mag, -10.f, 10.f, LEVELS);

        float jx = 0.f, jy = 0.f, jz = 0.f;
        if (n > 0) {
            float dt = inp[n * 4 + 0] - inp[(n - 1) * 4 + 0];
            jx = (x - inp[(n - 1) * 4 + 1]) / dt;
            jy = (y - inp[(n - 1) * 4 + 2]) / dt;
            jz = (z - inp[(n - 1) * 4 + 3]) / dt;
        }
        int ijx = lvl_idx(jx, -5.f, 5.f, LEVELS);
        int ijy = lvl_idx(jy, -5.f, 5.f, LEVELS);
        int ijz = lvl_idx(jz, -5.f, 5.f, LEVELS);
        float jmag = sqrtf(jx * jx + jy * jy + jz * jz);
        int ijm = lvl_idx(jmag, -10.f, 10.f, LEVELS);
        int it = lvl_idx(tn, 0.f, (float)TSTAMPS, TSTAMPS);

        int* p = sidx + n * 9;
        p[0] = ix * DIM;  p[1] = iy * DIM;  p[2] = iz * DIM;
        p[3] = im * DIM;  p[4] = ijx * DIM; p[5] = ijy * DIM;
        p[6] = ijz * DIM; p[7] = ijm * DIM; p[8] = it * DIM;
    }
    red[0][tid] = ex; red[1][tid] = ey; red[2][tid] = ez;
    __syncthreads();
    for (int s = 128; s > 0; s >>= 1) {
        if (tid < s) {
            red[0][tid] += red[0][tid + s];
            red[1][tid] += red[1][tid + s];
            red[2][tid] += red[2][tid + s];
        }
        __syncthreads();
    }
    if (tid == 0) {
        const float inv = 1.0f / (float)NSAMP;
        eidx[0] = lvl_idx(red[0][0] * inv, -10.f, 10.f, LEVELS) * DIM;
        eidx[1] = lvl_idx(red[1][0] * inv, -10.f, 10.f, LEVELS) * DIM;
        eidx[2] = lvl_idx(red[2][0] * inv, -10.f, 10.f, LEVELS) * DIM;
    }
}

// ---------------- Kernel 2: gather-product accumulate (L2-bandwidth bound) ----
__global__ void __launch_bounds__(256)
hdc_accum(const float* __restrict__ Wx,  const float* __restrict__ Wy,
          const float* __restrict__ Wz,  const float* __restrict__ Wm,
          const float* __restrict__ Wjx, const float* __restrict__ Wjy,
          const float* __restrict__ Wjz, const float* __restrict__ Wjm,
          const float* __restrict__ Wt,
          const int* __restrict__ sidx, float* __restrict__ partial) {
    __shared__ int sb[SPS * 9];
    const int tid = threadIdx.x;
    const int slice = blockIdx.y;
    const int n0 = slice * SPS;
    for (int i = tid; i < SPS * 9; i += 256) sb[i] = sidx[n0 * 9 + i];
    __syncthreads();

    const int c = blockIdx.x * 256 + tid;       // float4 column group
    if (c >= DIM / 4) return;
    const int col = c * 4;

    float4 acc = make_float4(0.f, 0.f, 0.f, 0.f);
    const int* sp = sb;
    for (int n = 0; n < SPS; ++n, sp += 9) {
        int b0 = __builtin_amdgcn_readfirstlane(sp[0]);
        int b1 = __builtin_amdgcn_readfirstlane(sp[1]);
        int b2 = __builtin_amdgcn_readfirstlane(sp[2]);
        int b3 = __builtin_amdgcn_readfirstlane(sp[3]);
        int b4 = __builtin_amdgcn_readfirstlane(sp[4]);
        int b5 = __builtin_amdgcn_readfirstlane(sp[5]);
        int b6 = __builtin_amdgcn_readfirstlane(sp[6]);
        int b7 = __builtin_amdgcn_readfirstlane(sp[7]);
        int b8 = __builtin_amdgcn_readfirstlane(sp[8]);

        if (n + 2 < SPS) {                      // gfx1250 global_prefetch_b8
            const int* pf = sp + 18;
            __builtin_prefetch(&Wx[pf[0] + col], 0, 3);
            __builtin_prefetch(&Wy[pf[1] + col], 0, 3);
            __builtin_prefetch(&Wz[pf[2] + col], 0, 3);
            __builtin_prefetch(&Wm[pf[3] + col], 0, 3);
            __builtin_prefetch(&Wjx[pf[4] + col], 0, 3);
            __builtin_prefetch(&Wjy[pf[5] + col], 0, 3);
            __builtin_prefetch(&Wjz[pf[6] + col], 0, 3);
            __builtin_prefetch(&Wjm[pf[7] + col], 0, 3);
            __builtin_prefetch(&Wt[pf[8] + col], 0, 3);
        }

        float4 v0 = *(const float4*)(Wx + b0 + col);
        float4 v1 = *(const float4*)(Wy + b1 + col);
        float4 v2 = *(const float4*)(Wz + b2 + col);
        float4 v3 = *(const float4*)(Wm + b3 + col);
        float4 v4 = *(const float4*)(Wjx + b4 + col);
        float4 v5 = *(const float4*)(Wjy + b5 + col);
        float4 v6 = *(const float4*)(Wjz + b6 + col);
        float4 v7 = *(const float4*)(Wjm + b7 + col);
        float4 v8 = *(const float4*)(Wt + b8 + col);

        acc.x += v0.x * v1.x * v2.x * v3.x * v4.x * v5.x * v6.x * v7.x * v8.x;
        acc.y += v0.y * v1.y * v2.y * v3.y * v4.y * v5.y * v6.y * v7.y * v8.y;
        acc.z += v0.z * v1.z * v2.z * v3.z * v4.z * v5.z * v6.z * v7.z * v8.z;
        acc.w += v0.w * v1.w * v2.w * v3.w * v4.w * v5.w * v6.w * v7.w * v8.w;
    }
    *(float4*)(partial + slice * DIM + col) = acc;
}

// ---------------- Kernel 3: WMMA cross-slice reduction + energy scale + tanh --
// Σₛ partial[s][d] over 16 slices == ones(16x4) @ B(4x16), accumulated over 4
// WMMA calls (16 K-values total). A = all-ones so result is K-order invariant;
// D row 0 (VGPR0, lanes: N = lane&15) carries the per-column sum.
__global__ void __launch_bounds__(128)
hdc_reduce(const float* __restrict__ partial,
           const float* __restrict__ Wex, const float* __restrict__ Wey,
           const float* __restrict__ Wez, const int* __restrict__ eidx,
           float* __restrict__ out) {
    const int tid = threadIdx.x;
    const int wave = tid >> 5;
    const int lane = tid & 31;
    const int colbase = (blockIdx.x * 4 + wave) * 16;
    const int mycol = colbase + (lane & 15);
    const int colc = mycol < DIM ? mycol : DIM - 1;   // clamp loads, guard store
    const int h = (lane >> 4) & 1;

    v2f a; a.x = 1.0f; a.y = 1.0f;                    // ones 16x4 A-matrix
    v8f acc = {};
#pragma unroll
    for (int j = 0; j < 4; ++j) {
        const int s0 = j * 4 + h * 2;                 // covers slices 0..15
        v2f b;
        b.x = partial[(s0 + 0) * DIM + colc];
        b.y = partial[(s0 + 1) * DIM + colc];
        acc = __builtin_amdgcn_wmma_f32_16x16x4_f32(
            /*neg_a=*/false, a, /*neg_b=*/false, b,
            /*c_mod=*/(short)0, acc, /*reuse_a=*/false, /*reuse_b=*/false);
    }
    float sum = acc[0];                               // row 0 == column sum
    float e = Wex[eidx[0] + colc] * Wey[eidx[1] + colc] * Wez[eidx[2] + colc];
    float val = tanhf(sum * e);
    if (lane < 16 && mycol < DIM) out[mycol] = val;
}

extern "C" void kernel_launch(void* const* d_in, const int* in_sizes, int n_in,
                              void* d_out, int out_size, void* d_ws, size_t ws_size,
                              hipStream_t stream) {
    const float* inp = (const float*)d_in[0];
    const float* Wx  = (const float*)d_in[1];
    const float* Wy  = (const float*)d_in[2];
    const float* Wz  = (const float*)d_in[3];
    const float* Wm  = (const float*)d_in[4];
    const float* Wjx = (const float*)d_in[5];
    const float* Wjy = (const float*)d_in[6];
    const float* Wjz = (const float*)d_in[7];
    const float* Wjm = (const float*)d_in[8];
    const float* Wex = (const float*)d_in[9];
    const float* Wey = (const float*)d_in[10];
    const float* Wez = (const float*)d_in[11];
    const float* Wt  = (const float*)d_in[12];

    int*   sidx    = (int*)((char*)d_ws + SIDX_OFF);
    int*   eidx    = (int*)((char*)d_ws + EIDX_OFF);
    float* partial = (float*)((char*)d_ws + PARTIAL_OFF);
    float* out     = (float*)d_out;

    hdc_prep<<<1, 256, 0, stream>>>(inp, sidx, eidx);

    dim3 g2((DIM / 4 + 255) / 256, NSLICE);   // (10, 16)
    hdc_accum<<<g2, 256, 0, stream>>>(Wx, Wy, Wz, Wm, Wjx, Wjy, Wjz, Wjm, Wt,
                                      sidx, partial);

    int g3 = (DIM + 63) / 64;                 // 157 blocks, 4 waves * 16 cols
    hdc_reduce<<<g3, 128, 0, stream>>>(partial, Wex, Wey, Wez, eidx, out);
}